// _AdaptivePool_31318901523122
// MI455X (gfx1250) — compile-verified
//
#include <hip/hip_runtime.h>
#include <stdint.h>

// Problem constants (from the reference): x is (32, 250, 250, 128) f32 NHWC,
// output (32, 7, 7, 128). 250 -> 7 adaptive windows: nb=35, rem=5.
#define B_ 32
#define H_ 250
#define W_ 250
#define C_ 128
#define T_ 7
#define MAXWIN 36
#define CHUNK_ROWS 2
#define CHUNK_FLOATS (CHUNK_ROWS * MAXWIN * C_)   // 9216 floats = 36864 B per buffer

typedef __attribute__((ext_vector_type(4))) uint32_t u32x4;
typedef __attribute__((ext_vector_type(8))) uint32_t u32x8;

// Issue one TDM 2D tile load: rows x row_elems f32 elements, row stride W_*C_
// elements in global memory, packed contiguously into LDS at lds_byte_off.
// D# packing per CDNA5 ISA 8.3/8.4 (2-group form, VADDR2/3 = NULL -> <=2D).
__device__ __forceinline__ void tdm_load_2d(uint64_t gaddr_bytes,
                                            uint32_t lds_byte_off,
                                            int rows, int row_elems) {
  u32x4 g0;
  g0[0] = 1u;                                            // count=1, user mode
  g0[1] = lds_byte_off;                                  // lds_addr [63:32]
  g0[2] = (uint32_t)gaddr_bytes;                         // global_addr[31:0]
  g0[3] = ((uint32_t)(gaddr_bytes >> 32) & 0x01ffffffu)  // global_addr[56:32]
          | 0x80000000u;                                 // type=2 ("image")
  u32x8 g1;
  g1[0] = 0x00020000u;                       // workgroup_mask=0, data_size=2 (4B)
  g1[1] = ((uint32_t)(W_ * C_) & 0xffffu) << 16;         // tensor_dim0 lo16 @ [63:48]
  g1[2] = ((uint32_t)(W_ * C_) >> 16)                    // tensor_dim0 hi16 @ [79:64]
          | (256u << 16);                                // tensor_dim1 lo16 @ [111:80]
  g1[3] = (uint32_t)row_elems << 16;                     // tile_dim0 @ [127:112]
  g1[4] = (uint32_t)rows;                                // tile_dim1 @ [143:128]
  g1[5] = (uint32_t)(W_ * C_);                           // tensor_dim0_stride lo32
  g1[6] = 0u;                                            // stride hi / dim1_stride lo
  g1[7] = 0u;
  asm volatile("tensor_load_to_lds %0, %1" :: "s"(g0), "s"(g1) : "memory");
}

__global__ __launch_bounds__(256) void
_AdaptivePool_31318901523122_kernel(const float* __restrict__ x,
                                    float* __restrict__ out) {
  __shared__ float smem[2 * CHUNK_FLOATS];   // double-buffered TDM landing zone

  const int tid  = threadIdx.x;
  const int c4   = tid & 31;    // channel quad: channels [4*c4, 4*c4+3]
  const int part = tid >> 5;    // pixel-partition id (8 waves)

  const int idx = blockIdx.x;           // one block per output pixel
  const int b   = idx / (T_ * T_);
  const int rr  = idx % (T_ * T_);
  const int oh  = rr / T_;
  const int ow  = rr % T_;

  // adaptive window geometry (250 -> 7): first 5 bins are width 36, last 2 are 35
  const int h0 = oh * 35 + (oh < 5 ? oh : 5);
  const int wh = 35 + (oh < 5 ? 1 : 0);
  const int w0 = ow * 35 + (ow < 5 ? ow : 5);
  const int ww = 35 + (ow < 5 ? 1 : 0);

  const int row_elems = ww * C_;                       // contiguous floats per window row
  const int nchunks   = (wh + CHUNK_ROWS - 1) / CHUNK_ROWS;

  const uint64_t base =
      (uint64_t)(uintptr_t)x +
      4ull * (uint64_t)(((b * H_ + h0) * W_ + w0)) * (uint64_t)C_;
  const uint32_t lds_base = (uint32_t)(uintptr_t)(void*)smem;   // low 32 bits = LDS offset

  const int wid = __builtin_amdgcn_readfirstlane(tid >> 5);     // scalar wave id

  // prologue: wave 0 issues chunk 0 into buffer 0
  if (wid == 0) {
    const int rows0 = (wh < CHUNK_ROWS) ? wh : CHUNK_ROWS;
    tdm_load_2d(base, lds_base, rows0, row_elems);
  }

  float ax = 0.f, ay = 0.f, az = 0.f, aw = 0.f;
  int buf = 0;
  for (int k = 0; k < nchunks; ++k) {
    const int rem  = wh - k * CHUNK_ROWS;
    const int rows = (rem < CHUNK_ROWS) ? rem : CHUNK_ROWS;

    if (wid == 0) {
      if (k + 1 < nchunks) {
        const int rem1  = wh - (k + 1) * CHUNK_ROWS;
        const int rows1 = (rem1 < CHUNK_ROWS) ? rem1 : CHUNK_ROWS;
        tdm_load_2d(base + 4ull * (uint64_t)(k + 1) * CHUNK_ROWS * (W_ * C_),
                    lds_base + (uint32_t)((buf ^ 1) * (CHUNK_FLOATS * 4)),
                    rows1, row_elems);
        __builtin_amdgcn_s_wait_tensorcnt(1);   // in-order: chunk k has landed
      } else {
        __builtin_amdgcn_s_wait_tensorcnt(0);
      }
    }
    __syncthreads();                            // chunk k visible to all waves

    const float4* src = (const float4*)&smem[buf * CHUNK_FLOATS];
    const int pixels = rows * ww;
    for (int p = part; p < pixels; p += 8) {    // lanes 0..31 read 512B: conflict-free
      float4 v = src[p * 32 + c4];
      ax += v.x; ay += v.y; az += v.z; aw += v.w;
    }
    __syncthreads();                            // all reads of buf done before overwrite
    buf ^= 1;
  }

  // cross-wave reduction of the 8 partial sums per channel quad
  float4* red = (float4*)smem;
  red[tid] = make_float4(ax, ay, az, aw);
  __syncthreads();
  if (tid < 32) {
    float sx = 0.f, sy = 0.f, sz = 0.f, sw = 0.f;
#pragma unroll
    for (int i = 0; i < 8; ++i) {
      float4 v = red[i * 32 + tid];
      sx += v.x; sy += v.y; sz += v.z; sw += v.w;
    }
    const float scale = 1.0f / (float)(wh * ww);
    float4* outv = (float4*)out;
    outv[((b * T_ + oh) * T_ + ow) * 32 + tid] =
        make_float4(sx * scale, sy * scale, sz * scale, sw * scale);
  }
}

extern "C" void kernel_launch(void* const* d_in, const int* in_sizes, int n_in,
                              void* d_out, int out_size, void* d_ws, size_t ws_size,
                              hipStream_t stream) {
  (void)in_sizes; (void)n_in; (void)d_ws; (void)ws_size; (void)out_size;
  const float* x = (const float*)d_in[0];
  float* out = (float*)d_out;
  _AdaptivePool_31318901523122_kernel<<<dim3(B_ * T_ * T_), dim3(256), 0, stream>>>(x, out);
}